// GXformer_58162447123012
// MI455X (gfx1250) — compile-verified
//
#include <hip/hip_runtime.h>
#include <math.h>

#define NPTS 16384   // N
#define NB   4       // B
#define MS   2048    // M = N/R
#define KG   16
#define CIN  32
#define COUT 32
#define GT_C 96

typedef float v2f __attribute__((ext_vector_type(2)));
typedef float v8f __attribute__((ext_vector_type(8)));

__device__ __forceinline__ float wave_sum32(float v) {
#pragma unroll
  for (int off = 1; off < 32; off <<= 1) v += __shfl_xor(v, off, 32);
  return v;
}

// Async global->LDS 16-byte copy (CDNA5 GLOBAL_LOAD_ASYNC_TO_LDS_B128,
// tracked by ASYNCcnt). lds_byte_off = LDS offset, gaddr = 64-bit global addr.
__device__ __forceinline__ void async_copy_b128(unsigned lds_byte_off,
                                                unsigned long long gaddr) {
  asm volatile("global_load_async_to_lds_b128 %0, %1, off"
               :: "v"(lds_byte_off), "v"(gaddr)
               : "memory");
}
__device__ __forceinline__ void async_wait0() {
  asm volatile("s_wait_asynccnt 0" ::: "memory");
}
__device__ __forceinline__ unsigned lds_offset_of(const void* p) {
  // Generic shared-aperture pointer: low 32 bits are the LDS byte offset.
  return (unsigned)(unsigned long long)p;
}

// ---------------------------------------------------------------------------
// 1) Farthest point sampling: one workgroup per batch, full point set in LDS
//    (196KB — uses CDNA5's 320KB-per-WGP LDS).
// ---------------------------------------------------------------------------
__global__ void fps_kernel(const float* __restrict__ xytp, int* __restrict__ out) {
  extern __shared__ float sm[];
  float* sx = sm;
  float* sy = sm + NPTS;
  float* st = sm + 2 * NPTS;
  __shared__ float rbv[32];
  __shared__ int   rbi[32];
  __shared__ int   s_last;

  const int b   = blockIdx.x;
  const int tid = threadIdx.x;           // 1024 threads
  const int lane = tid & 31, wid = tid >> 5;

  const float4* xy4 = (const float4*)xytp + (size_t)b * NPTS;
  for (int p = tid; p < NPTS; p += 1024) {
    float4 v = xy4[p];
    sx[p] = v.x; sy[p] = v.y; st[p] = v.z;
  }
  if (tid == 0) s_last = 0;
  __syncthreads();

  float px[16], py[16], pt[16], mind[16];
#pragma unroll
  for (int j = 0; j < 16; ++j) {
    int p = tid + j * 1024;
    px[j] = sx[p]; py[j] = sy[p]; pt[j] = st[p];
    mind[j] = 1e10f;
  }

  int last = 0;
  for (int i = 0; i < MS; ++i) {
    if (tid == 0) out[b * MS + i] = last;   // scan emits *previous* last
    const float lx = sx[last], ly = sy[last], lt = st[last];
    float best = -1.0f; int bi = 0x7fffffff;
#pragma unroll
    for (int j = 0; j < 16; ++j) {
      float dx = px[j] - lx, dy = py[j] - ly, dt = pt[j] - lt;
      float d = dx * dx + dy * dy + dt * dt;
      mind[j] = fminf(mind[j], d);
      if (mind[j] > best) { best = mind[j]; bi = tid + j * 1024; }
    }
#pragma unroll
    for (int off = 16; off > 0; off >>= 1) {
      float ov = __shfl_xor(best, off, 32);
      int   oi = __shfl_xor(bi, off, 32);
      if (ov > best || (ov == best && oi < bi)) { best = ov; bi = oi; }
    }
    if (lane == 0) { rbv[wid] = best; rbi[wid] = bi; }
    __syncthreads();
    if (wid == 0) {
      float bv = rbv[lane]; int bx = rbi[lane];
#pragma unroll
      for (int off = 16; off > 0; off >>= 1) {
        float ov = __shfl_xor(bv, off, 32);
        int   oi = __shfl_xor(bx, off, 32);
        if (ov > bv || (ov == bv && oi < bx)) { bv = ov; bx = oi; }
      }
      if (lane == 0) s_last = bx;
    }
    __syncthreads();
    last = s_last;
  }
}

// ---------------------------------------------------------------------------
// 2a) kNN: for each of M samples, top-16 nearest among all N points.
//     Candidate set staged into LDS as AoS float4 (256KB) with CDNA5 async
//     global->LDS b128 copies; inner loop reads are broadcast ds_load_b128.
// ---------------------------------------------------------------------------
__global__ void knn_pair_kernel(const float* __restrict__ xytp,
                                const int* __restrict__ sample_idx,
                                int* __restrict__ pair_idx) {
  extern __shared__ float4 sm4[];      // NPTS float4
  const int b = blockIdx.x >> 3;
  const int g = blockIdx.x & 7;
  const int tid = threadIdx.x;  // 256

  const float4* xy4 = (const float4*)xytp + (size_t)b * NPTS;
  const unsigned lbase = lds_offset_of(sm4);
  for (int p = tid; p < NPTS; p += 256)
    async_copy_b128(lbase + (unsigned)p * 16u,
                    (unsigned long long)(const void*)(xy4 + p));
  async_wait0();
  __syncthreads();

  const int m = g * 256 + tid;
  const int qi = sample_idx[b * MS + m];
  const float4 qv = sm4[qi];

  float bd[KG]; int bix[KG];
#pragma unroll
  for (int j = 0; j < KG; ++j) { bd[j] = 3.4e38f; bix[j] = -1; }

  for (int p = 0; p < NPTS; ++p) {
    float4 v = sm4[p];
    float dx = v.x - qv.x, dy = v.y - qv.y, dt = v.z - qv.z;
    float d = dx * dx + dy * dy + dt * dt;
    if (d < bd[KG - 1]) {
      bd[KG - 1] = d; bix[KG - 1] = p;
#pragma unroll
      for (int j = KG - 1; j > 0; --j) {
        if (bd[j] < bd[j - 1]) {
          float t = bd[j]; bd[j] = bd[j - 1]; bd[j - 1] = t;
          int ti = bix[j]; bix[j] = bix[j - 1]; bix[j - 1] = ti;
        }
      }
    }
  }
#pragma unroll
  for (int j = 0; j < KG; ++j) pair_idx[(b * MS + m) * KG + j] = bix[j];
}

// ---------------------------------------------------------------------------
// 2b) inverse kNN: for each of N points, top-16 nearest among M samples.
//     Gathered sample coords staged to LDS via per-lane async b128 copies.
// ---------------------------------------------------------------------------
__global__ void knn_inv_kernel(const float* __restrict__ xytp,
                               const int* __restrict__ sample_idx,
                               int* __restrict__ inv_idx) {
  extern __shared__ float4 sm4[];      // MS float4
  const int b = blockIdx.x >> 6;
  const int g = blockIdx.x & 63;
  const int tid = threadIdx.x;

  const float4* xy4 = (const float4*)xytp + (size_t)b * NPTS;
  const unsigned lbase = lds_offset_of(sm4);
  for (int s = tid; s < MS; s += 256) {
    int si = sample_idx[b * MS + s];
    async_copy_b128(lbase + (unsigned)s * 16u,
                    (unsigned long long)(const void*)(xy4 + si));
  }
  async_wait0();
  __syncthreads();

  const int n = g * 256 + tid;
  const float4 q = xy4[n];

  float bd[KG]; int bix[KG];
#pragma unroll
  for (int j = 0; j < KG; ++j) { bd[j] = 3.4e38f; bix[j] = -1; }

  for (int s = 0; s < MS; ++s) {
    float4 v = sm4[s];
    float dx = v.x - q.x, dy = v.y - q.y, dt = v.z - q.z;
    float d = dx * dx + dy * dy + dt * dt;
    if (d < bd[KG - 1]) {
      bd[KG - 1] = d; bix[KG - 1] = s;
#pragma unroll
      for (int j = KG - 1; j > 0; --j) {
        if (bd[j] < bd[j - 1]) {
          float t = bd[j]; bd[j] = bd[j - 1]; bd[j - 1] = t;
          int ti = bix[j]; bix[j] = bix[j - 1]; bix[j - 1] = ti;
        }
      }
    }
  }
#pragma unroll
  for (int j = 0; j < KG; ++j) inv_idx[((size_t)b * NPTS + n) * KG + j] = bix[j];
}

// ---------------------------------------------------------------------------
// 3) gt = features @ wg + bg via V_WMMA_F32_16X16X4_F32.
//    One wave -> one 16x16 output tile, 8 WMMA steps over K=32.
// ---------------------------------------------------------------------------
__global__ void gemm_gt_kernel(const float* __restrict__ feat,
                               const float* __restrict__ wg,
                               const float* __restrict__ bg,
                               float* __restrict__ gt) {
  const int wave = (blockIdx.x * blockDim.x + threadIdx.x) >> 5;
  const int lane = threadIdx.x & 31;
  const int rt    = wave & 4095;        // row tile (65536/16 = 4096)
  const int ctile = wave >> 12;         // col tile (96/16 = 6)
  const int row0 = rt * 16;
  const int c0 = ctile * 16;
  const int l15 = lane & 15;
  const int ksel = (lane >> 4) << 1;    // 0 or 2
  const int mrow = row0 + l15;

  v8f acc = {0.f, 0.f, 0.f, 0.f, 0.f, 0.f, 0.f, 0.f};
#pragma unroll
  for (int k0 = 0; k0 < CIN; k0 += 4) {
    v2f a, bm;
    a[0]  = feat[(size_t)mrow * CIN + k0 + ksel];
    a[1]  = feat[(size_t)mrow * CIN + k0 + ksel + 1];
    bm[0] = wg[(k0 + ksel) * GT_C + c0 + l15];
    bm[1] = wg[(k0 + ksel + 1) * GT_C + c0 + l15];
    acc = __builtin_amdgcn_wmma_f32_16x16x4_f32(false, a, false, bm,
                                                (short)0, acc, false, false);
  }
  const int ccol = c0 + l15;
  const float bgv = bg[ccol];
  const int rbase = row0 + ((lane >> 4) << 3);
#pragma unroll
  for (int r = 0; r < 8; ++r)
    gt[(size_t)(rbase + r) * GT_C + ccol] = acc[r] + bgv;
}

// ---------------------------------------------------------------------------
// 4) neighbor max-pool of psi/alpha over pair_idx; also gather sample_xytp.
// ---------------------------------------------------------------------------
__global__ void pool_kernel(const float* __restrict__ gt,
                            const int* __restrict__ pair_idx,
                            const int* __restrict__ sample_idx,
                            const float* __restrict__ xytp,
                            float* __restrict__ psi_m,
                            float* __restrict__ alpha_m,
                            float* __restrict__ samp_xytp) {
  const int t = blockIdx.x * 256 + threadIdx.x;   // NB*MS*64 threads
  const int c64 = t & 63;
  const int bm = t >> 6;          // b*MS + m
  const int b = bm >> 11;
  const int chan = c64 & 31;
  const int off = (c64 < 32) ? (COUT + chan) : (2 * COUT + chan);
  float mx = -3.4e38f;
#pragma unroll
  for (int k = 0; k < KG; ++k) {
    int p = pair_idx[bm * KG + k];
    float v = gt[((size_t)b * NPTS + p) * GT_C + off];
    mx = fmaxf(mx, v);
  }
  if (c64 < 32) psi_m[bm * COUT + chan] = mx;
  else          alpha_m[bm * COUT + chan] = mx;
  if (c64 < 4) {
    int si = sample_idx[bm];
    samp_xytp[bm * 4 + c64] = xytp[((size_t)b * NPTS + si) * 4 + c64];
  }
}

// ---------------------------------------------------------------------------
// 5) BN statistics: mean + covariance of rel (h is affine in rel, so BN
//    folds into 14 accumulated scalars instead of materializing h).
// ---------------------------------------------------------------------------
__global__ void stats_kernel(const float* __restrict__ xytp,
                             const int* __restrict__ inv_idx,
                             const float* __restrict__ samp_xytp,
                             float* __restrict__ stats) {
  const int gs = gridDim.x * blockDim.x;
  const int lane = threadIdx.x & 31;
  float a0 = 0, a1 = 0, a2 = 0, a3 = 0;
  float s00 = 0, s01 = 0, s02 = 0, s03 = 0, s11 = 0, s12 = 0, s13 = 0;
  float s22 = 0, s23 = 0, s33 = 0;
  const int total = NB * NPTS * KG;
  for (int e = blockIdx.x * blockDim.x + threadIdx.x; e < total; e += gs) {
    int k = e & 15;
    int bn = e >> 4;            // b*N + n
    int b = bn >> 14;
    int m = inv_idx[(size_t)bn * KG + k];
    float4 q = ((const float4*)xytp)[bn];
    float4 s = ((const float4*)samp_xytp)[b * MS + m];
    float r0 = q.x - s.x, r1 = q.y - s.y, r2 = q.z - s.z, r3 = q.w - s.w;
    a0 += r0; a1 += r1; a2 += r2; a3 += r3;
    s00 += r0 * r0; s01 += r0 * r1; s02 += r0 * r2; s03 += r0 * r3;
    s11 += r1 * r1; s12 += r1 * r2; s13 += r1 * r3;
    s22 += r2 * r2; s23 += r2 * r3; s33 += r3 * r3;
  }
  a0 = wave_sum32(a0);  a1 = wave_sum32(a1);  a2 = wave_sum32(a2);  a3 = wave_sum32(a3);
  s00 = wave_sum32(s00); s01 = wave_sum32(s01); s02 = wave_sum32(s02); s03 = wave_sum32(s03);
  s11 = wave_sum32(s11); s12 = wave_sum32(s12); s13 = wave_sum32(s13);
  s22 = wave_sum32(s22); s23 = wave_sum32(s23); s33 = wave_sum32(s33);
  __shared__ float ls[14];
  if (threadIdx.x < 14) ls[threadIdx.x] = 0.f;
  __syncthreads();
  if (lane == 0) {
    atomicAdd(&ls[0], a0);  atomicAdd(&ls[1], a1);  atomicAdd(&ls[2], a2);  atomicAdd(&ls[3], a3);
    atomicAdd(&ls[4], s00); atomicAdd(&ls[5], s01); atomicAdd(&ls[6], s02); atomicAdd(&ls[7], s03);
    atomicAdd(&ls[8], s11); atomicAdd(&ls[9], s12); atomicAdd(&ls[10], s13);
    atomicAdd(&ls[11], s22); atomicAdd(&ls[12], s23); atomicAdd(&ls[13], s33);
  }
  __syncthreads();
  if (threadIdx.x < 14) atomicAdd(&stats[threadIdx.x], ls[threadIdx.x]);
}

__global__ void bn_finalize_kernel(const float* __restrict__ stats,
                                   const float* __restrict__ w1,
                                   const float* __restrict__ b1,
                                   const float* __restrict__ bn_g,
                                   const float* __restrict__ bn_b,
                                   float* __restrict__ bnAD) {
  const int c = threadIdx.x;
  if (c >= 4) return;
  const float inv = 1.0f / (float)(NB * NPTS * KG);
  float m0 = stats[0] * inv, m1 = stats[1] * inv, m2 = stats[2] * inv, m3 = stats[3] * inv;
  float C00 = stats[4] * inv - m0 * m0, C01 = stats[5] * inv - m0 * m1;
  float C02 = stats[6] * inv - m0 * m2, C03 = stats[7] * inv - m0 * m3;
  float C11 = stats[8] * inv - m1 * m1, C12 = stats[9] * inv - m1 * m2;
  float C13 = stats[10] * inv - m1 * m3, C22 = stats[11] * inv - m2 * m2;
  float C23 = stats[12] * inv - m2 * m3, C33 = stats[13] * inv - m3 * m3;
  float u0 = w1[0 * 4 + c], u1 = w1[1 * 4 + c], u2 = w1[2 * 4 + c], u3 = w1[3 * 4 + c];
  float mu = b1[c] + m0 * u0 + m1 * u1 + m2 * u2 + m3 * u3;
  float var = u0 * u0 * C00 + u1 * u1 * C11 + u2 * u2 * C22 + u3 * u3 * C33 +
              2.f * (u0 * u1 * C01 + u0 * u2 * C02 + u0 * u3 * C03 +
                     u1 * u2 * C12 + u1 * u3 * C13 + u2 * u3 * C23);
  float a = bn_g[c] * rsqrtf(var + 1e-5f);
  bnAD[c] = a;
  bnAD[4 + c] = bn_b[c] - mu * a;
}

// ---------------------------------------------------------------------------
// 6) Fused attention: one wave per point, lane = output channel (C_OUT == 32
//    == wave32). LayerNorm = cross-lane butterfly; softmax over K per-lane.
// ---------------------------------------------------------------------------
__global__ void attn_kernel(const float* __restrict__ xytp,
                            const int* __restrict__ inv_idx,
                            const float* __restrict__ samp_xytp,
                            const float* __restrict__ gt,
                            const float* __restrict__ psi_m,
                            const float* __restrict__ alpha_m,
                            const float* __restrict__ w1,
                            const float* __restrict__ b1,
                            const float* __restrict__ w2,
                            const float* __restrict__ b2,
                            const float* __restrict__ ln_g,
                            const float* __restrict__ ln_b,
                            const float* __restrict__ bnAD,
                            float* __restrict__ out) {
  const int lane = threadIdx.x & 31;
  const int bn = (blockIdx.x * blockDim.x + threadIdx.x) >> 5;  // b*N + n
  const int b = bn >> 14;
  const int c = lane;

  const float w2c0 = w2[0 * COUT + c], w2c1 = w2[1 * COUT + c];
  const float w2c2 = w2[2 * COUT + c], w2c3 = w2[3 * COUT + c];
  const float b2c = b2[c];
  const float lng = ln_g[c], lnb = ln_b[c];
  const float A0 = bnAD[0], A1 = bnAD[1], A2 = bnAD[2], A3 = bnAD[3];
  const float D0 = bnAD[4], D1 = bnAD[5], D2 = bnAD[6], D3 = bnAD[7];
  const float b10 = b1[0], b11 = b1[1], b12 = b1[2], b13 = b1[3];
  float W[16];
#pragma unroll
  for (int i = 0; i < 16; ++i) W[i] = w1[i];

  const float4 q = ((const float4*)xytp)[bn];
  const float varphi = gt[(size_t)bn * GT_C + c];
  int idx_l = 0;
  if (lane < KG) idx_l = inv_idx[(size_t)bn * KG + lane];

  float logit[KG], val[KG];
  const float inv_scale = 0.17677669529663687f;  // 1/sqrt(32)
#pragma unroll
  for (int k = 0; k < KG; ++k) {
    int ik = __shfl(idx_l, k, 32);
    int smidx = b * MS + ik;
    float4 sp = ((const float4*)samp_xytp)[smidx];
    float r0 = q.x - sp.x, r1 = q.y - sp.y, r2 = q.z - sp.z, r3 = q.w - sp.w;
    float h0 = r0 * W[0] + r1 * W[4] + r2 * W[8]  + r3 * W[12] + b10;
    float h1 = r0 * W[1] + r1 * W[5] + r2 * W[9]  + r3 * W[13] + b11;
    float h2 = r0 * W[2] + r1 * W[6] + r2 * W[10] + r3 * W[14] + b12;
    float h3 = r0 * W[3] + r1 * W[7] + r2 * W[11] + r3 * W[15] + b13;
    h0 = fmaxf(h0 * A0 + D0, 0.f);
    h1 = fmaxf(h1 * A1 + D1, 0.f);
    h2 = fmaxf(h2 * A2 + D2, 0.f);
    h3 = fmaxf(h3 * A3 + D3, 0.f);
    float delta = h0 * w2c0 + h1 * w2c1 + h2 * w2c2 + h3 * w2c3 + b2c;
    float psn = psi_m[smidx * COUT + c];
    float aln = alpha_m[smidx * COUT + c];
    float pre = varphi - psn + delta;
    float mean = wave_sum32(pre) * (1.f / 32.f);
    float dv = pre - mean;
    float vv = wave_sum32(dv * dv) * (1.f / 32.f);
    float ln = dv * rsqrtf(vv + 1e-5f) * lng + lnb;
    logit[k] = ln * inv_scale;
    val[k] = aln + delta;
  }
  float mx = logit[0];
#pragma unroll
  for (int k = 1; k < KG; ++k) mx = fmaxf(mx, logit[k]);
  float se = 0.f, acc = 0.f;
#pragma unroll
  for (int k = 0; k < KG; ++k) {
    float e = __expf(logit[k] - mx);
    se += e;
    acc += e * val[k];
  }
  out[(size_t)bn * COUT + c] = acc / se;
}

// ---------------------------------------------------------------------------
extern "C" void kernel_launch(void* const* d_in, const int* in_sizes, int n_in,
                              void* d_out, int out_size, void* d_ws, size_t ws_size,
                              hipStream_t stream) {
  const float* xytp     = (const float*)d_in[0];
  const float* features = (const float*)d_in[1];
  const float* w1   = (const float*)d_in[2];
  const float* b1   = (const float*)d_in[3];
  const float* bn_g = (const float*)d_in[4];
  const float* bn_b = (const float*)d_in[5];
  const float* w2   = (const float*)d_in[6];
  const float* b2   = (const float*)d_in[7];
  const float* wg   = (const float*)d_in[8];
  const float* bg   = (const float*)d_in[9];
  const float* ln_g = (const float*)d_in[10];
  const float* ln_b = (const float*)d_in[11];
  (void)in_sizes; (void)n_in; (void)out_size; (void)ws_size;

  char* ws = (char*)d_ws;
  size_t off = 0;
  auto alloc = [&](size_t bytes) -> void* {
    void* p = ws + off;
    off = (off + bytes + 255) & ~(size_t)255;
    return p;
  };
  float* stats      = (float*)alloc(64);
  float* bnAD       = (float*)alloc(32);
  int*   sample_idx = (int*)alloc((size_t)NB * MS * sizeof(int));
  int*   pair_idx   = (int*)alloc((size_t)NB * MS * KG * sizeof(int));
  int*   inv_idx    = (int*)alloc((size_t)NB * NPTS * KG * sizeof(int));
  float* gt         = (float*)alloc((size_t)NB * NPTS * GT_C * sizeof(float));
  float* samp_xytp  = (float*)alloc((size_t)NB * MS * 4 * sizeof(float));
  float* psi_m      = (float*)alloc((size_t)NB * MS * COUT * sizeof(float));
  float* alpha_m    = (float*)alloc((size_t)NB * MS * COUT * sizeof(float));

  hipMemsetAsync(stats, 0, 64, stream);

  fps_kernel<<<NB, 1024, NPTS * 3 * sizeof(float), stream>>>(xytp, sample_idx);
  gemm_gt_kernel<<<3072, 256, 0, stream>>>(features, wg, bg, gt);
  knn_pair_kernel<<<NB * 8, 256, NPTS * sizeof(float4), stream>>>(xytp, sample_idx, pair_idx);
  knn_inv_kernel<<<NB * 64, 256, MS * sizeof(float4), stream>>>(xytp, sample_idx, inv_idx);
  pool_kernel<<<(NB * MS * 64) / 256, 256, 0, stream>>>(gt, pair_idx, sample_idx, xytp,
                                                        psi_m, alpha_m, samp_xytp);
  stats_kernel<<<2048, 256, 0, stream>>>(xytp, inv_idx, samp_xytp, stats);
  bn_finalize_kernel<<<1, 32, 0, stream>>>(stats, w1, b1, bn_g, bn_b, bnAD);
  attn_kernel<<<(NB * NPTS) / 8, 256, 0, stream>>>(xytp, inv_idx, samp_xytp, gt,
                                                   psi_m, alpha_m, w1, b1, w2, b2,
                                                   ln_g, ln_b, bnAD, (float*)d_out);
}